// ScatterBlock_41420664602705
// MI455X (gfx1250) — compile-verified
//
#include <hip/hip_runtime.h>
#include <stdint.h>

// Problem constants (match reference)
#define B_     2
#define C_     256
#define H_     384
#define W_     384
#define NACT   16384
#define GRID_  192
// OFFSET=(0,0), STRIDE=(2,2) hardcoded below (h = 2*iy, w = 2*ix)

#define NPLANES (B_ * C_)         // 512
#define PLANE_SZ (H_ * W_)        // 147456 floats

typedef float v4f __attribute__((ext_vector_type(4)));

// ---- helper: byte offset of an LDS object from the wave's LDS base ----
__device__ __forceinline__ unsigned lds_off_u32(void* p) {
    return (unsigned)(size_t)(__attribute__((address_space(3))) void*)p;
}

// ------------------------------------------------------------------
// Kernel 1a: init inverse map to -1
// ------------------------------------------------------------------
__global__ void init_map_kernel(int* __restrict__ map, int n) {
    int i = blockIdx.x * blockDim.x + threadIdx.x;
    if (i < n) map[i] = -1;
}

// ------------------------------------------------------------------
// Kernel 1b: invert the scatter: map[iy*192+ix] = k (max-k wins on dups;
// duplicate behavior is nondeterministic in the JAX reference)
// ------------------------------------------------------------------
__global__ void build_map_kernel(const int* __restrict__ idx, int* __restrict__ map) {
    int k = blockIdx.x * blockDim.x + threadIdx.x;
    if (k < NACT) {
        int iy = idx[2 * k + 0];
        int ix = idx[2 * k + 1];
        atomicMax(&map[iy * GRID_ + ix], k);
    }
}

// ------------------------------------------------------------------
// Kernel 2: fused copy + gather.
//   grid = (H/ROWS bands, B*C planes), block = 256 threads (8 wave32)
//   - async-DMA this plane's 64KB x-slab into LDS (GLOBAL_LOAD_ASYNC_TO_LDS_B128)
//   - stream the band: NT b128 load of orig, patch even rows from LDS via map,
//     NT b128 store to out. Every output byte written exactly once, coalesced.
// ------------------------------------------------------------------
#define ROWS         96
#define TPB          256
#define F4_PER_ROW   (W_ / 4)               // 96
#define F4_PER_BLOCK (ROWS * F4_PER_ROW)    // 9216
#define ITERS        (F4_PER_BLOCK / TPB)   // 36
#define XCHUNKS      (NACT * 4 / 16 / TPB)  // 16 b128 chunks per thread

__global__ __launch_bounds__(TPB) void scatter_fused_kernel(
    const float* __restrict__ x,
    const float* __restrict__ orig,
    const int*   __restrict__ map,
    float*       __restrict__ out)
{
    __shared__ float lds_x[NACT];           // 64 KB: this plane's x[:, plane, :]

    const int plane = blockIdx.y;           // b*C + c, 0..511
    const int band  = blockIdx.x;           // 0..3
    const int tid   = threadIdx.x;

    // ---- stage 1: async bulk copy of x[plane, 0:16384] into LDS ----
    {
        const unsigned long long gbase =
            (unsigned long long)(const void*)(x + (size_t)plane * NACT);
        const unsigned lds0 = lds_off_u32(&lds_x[0]);
        #pragma unroll
        for (int i = 0; i < XCHUNKS; ++i) {
            unsigned byte_off = (unsigned)(i * TPB + tid) * 16u;  // 0..65520
            unsigned lds_addr = lds0 + byte_off;
            // GVS form: mem = SGPR64 base + VGPR32 offset ; dest = LDS byte addr
            asm volatile("global_load_async_to_lds_b128 %0, %1, %2"
                         :: "v"(lds_addr), "v"(byte_off), "s"(gbase)
                         : "memory");
        }
        asm volatile("s_wait_asynccnt 0" ::: "memory");
    }
    __syncthreads();

    // ---- stage 2: stream the band ----
    const int    h0         = band * ROWS;
    const float* orig_plane = orig + (size_t)plane * PLANE_SZ;
    float*       out_plane  = out  + (size_t)plane * PLANE_SZ;

    #pragma unroll 4
    for (int it = 0; it < ITERS; ++it) {
        int vid = it * TPB + tid;            // 0..9215
        int r   = vid / F4_PER_ROW;          // row within band
        int c4  = vid - r * F4_PER_ROW;      // float4 column, 0..95
        int h   = h0 + r;
        size_t off = (size_t)h * W_ + (size_t)(c4 * 4);

        v4f val = __builtin_nontemporal_load((const v4f*)(orig_plane + off));

        if ((h & 1) == 0) {                  // stride-2 rows: only even h active
            int iy = h >> 1;
            const int* mrow = map + iy * GRID_;
            int k0 = mrow[2 * c4];           // w = 4*c4     (even)
            int k1 = mrow[2 * c4 + 1];       // w = 4*c4 + 2 (even)
            if (k0 >= 0) val.x = lds_x[k0];
            if (k1 >= 0) val.z = lds_x[k1];
        }
        __builtin_nontemporal_store(val, (v4f*)(out_plane + off));
    }
}

// ------------------------------------------------------------------
extern "C" void kernel_launch(void* const* d_in, const int* in_sizes, int n_in,
                              void* d_out, int out_size, void* d_ws, size_t ws_size,
                              hipStream_t stream) {
    const float* x    = (const float*)d_in[0];  // [B, C, NACT]
    const float* orig = (const float*)d_in[1];  // [B, C, H, W]
    const int*   idx  = (const int*)  d_in[2];  // [NACT, 2]
    float*       out  = (float*)d_out;          // [B, C, H, W]
    int*         map  = (int*)d_ws;             // [GRID_*GRID_] = 144 KB scratch

    const int mapn = GRID_ * GRID_;             // 36864
    init_map_kernel<<<(mapn + 255) / 256, 256, 0, stream>>>(map, mapn);
    build_map_kernel<<<(NACT + 255) / 256, 256, 0, stream>>>(idx, map);

    dim3 grid(H_ / ROWS, NPLANES);              // (4, 512)
    scatter_fused_kernel<<<grid, TPB, 0, stream>>>(x, orig, map, out);
}